// YoloLayer_56693568307438
// MI455X (gfx1250) — compile-verified
//
#include <hip/hip_runtime.h>
#include <cstdint>

// ---------------------------------------------------------------------------
// Fused YOLO box decode for MI455X (gfx1250).
//
//   input : (32, 3*85, 52, 52) f32 -> (slice=96, ch=85, spatial=2704)
//   output: (96, 2704, 85) f32      (per-slice transpose + per-channel math)
//
// Roofline: 176 MB traffic -> ~7.6 us @ 23.3 TB/s. Pure bandwidth problem.
// Structure (per 85x104 tile, one 256-thread block):
//   Phase 1: coalesced b128 global loads; transform in registers (only rows
//            0..4 pay sigmoid/exp); scatter ds_store_b32 into LDS in [p][c]
//            layout == exact output layout (packed stride 85).
//   Phase 2: GLOBAL_STORE_ASYNC_FROM_LDS_B128 — LDS image is byte-identical
//            to the output tile, so copy-out is a linear async DMA with no
//            VALU work (ASYNCcnt-tracked CDNA5 path).
// ---------------------------------------------------------------------------

#define NB 32
#define NA 3
#define NCH 85                          // 5 + 80 classes
#define NHW 52
#define SPATIAL (NHW * NHW)             // 2704
#define NSLICES (NB * NA)               // 96
#define TILE_W 104                      // spatial positions per tile; 2*52, 2704/104=26
#define TILES_PER_SLICE (SPATIAL / TILE_W)   // 26

// Pointee type of the async-LDS builtins (from hipcc's round-1 diagnostic):
//   '__attribute__((__vector_size__(4 * sizeof(int)))) int __device__ *'
typedef int async_b128_t __attribute__((vector_size(16)));

__device__ __forceinline__ void async_store_b128(void* g, uint32_t lds_off) {
#if __has_builtin(__builtin_amdgcn_global_store_async_from_lds_b128)
    __builtin_amdgcn_global_store_async_from_lds_b128(
        (__attribute__((address_space(1))) async_b128_t*)(uintptr_t)g,
        (__attribute__((address_space(3))) async_b128_t*)(uintptr_t)lds_off,
        /*imm offset*/ 0, /*cpol*/ 0);
#else
    asm volatile("global_store_async_from_lds_b128 %0, %1, off"
                 :: "v"(g), "v"(lds_off)
                 : "memory");
#endif
}

__device__ __forceinline__ float sigmoidf_(float v) {
    return 1.0f / (1.0f + expf(-v));
}

__global__ __launch_bounds__(256)
void yolo_decode_kernel(const float* __restrict__ in, float* __restrict__ out) {
    // [p][c] layout, packed stride NCH -> byte-identical to the output tile.
    __shared__ __align__(16) float tile[TILE_W * NCH];   // 8840 floats = 35,360 B

    const int tid   = threadIdx.x;
    const int slice = blockIdx.x / TILES_PER_SLICE;          // b*3 + a
    const int tIdx  = blockIdx.x - slice * TILES_PER_SLICE;  // tile within slice
    const int a     = slice % NA;                            // anchor index
    const int s0    = tIdx * TILE_W;                         // s0 % 52 == 0
    const int gy0   = s0 / NHW;                              // grid_y of first row

    const float aw = (a == 0) ? 10.0f : ((a == 1) ? 16.0f : 33.0f);
    const float ah = (a == 0) ? 13.0f : ((a == 1) ? 30.0f : 23.0f);
    constexpr float INV_IMG = 1.0f / 416.0f;

    const float* src = in + (size_t)slice * (size_t)(NCH * SPATIAL) + s0;

    // ---- Phase 1: load (coalesced along spatial), transform, LDS-transpose -
    constexpr int VPR  = TILE_W / 4;        // 26 float4 per channel row
    constexpr int NVEC = VPR * NCH;         // 2210 lane-transfers per tile
    for (int i = tid; i < NVEC; i += 256) {
        const int c  = i / VPR;             // channel 0..84
        const int v4 = i - c * VPR;         // float4 index in row
        const int p0 = v4 * 4;              // local spatial position of lane's group

        const float4 v = *(const float4*)(src + (size_t)c * SPATIAL + p0);
        float r[4] = {v.x, v.y, v.z, v.w};

        if (c < 5) {                         // only 5/85 rows pay transcendentals
            #pragma unroll
            for (int k = 0; k < 4; ++k) {
                const int   p  = p0 + k;            // 0..103 -> spans 2 grid rows
                const int   hi = (p >= NHW) ? 1 : 0;
                const float gx = (float)(p - NHW * hi);
                const float gy = (float)(gy0 + hi);
                const float t  = r[k];
                if      (c == 0) r[k] = sigmoidf_(t) + gx;          // bx
                else if (c == 1) r[k] = sigmoidf_(t) + gy;          // by
                else if (c == 2) r[k] = expf(t * INV_IMG) * aw;     // bw
                else if (c == 3) r[k] = expf(t * INV_IMG) * ah;     // bh
                else             r[k] = sigmoidf_(t);               // conf
            }
        }

        // Transposing scatter: stride 4*85=340 floats/lane -> bank step 20,
        // gcd(20,64)=4 -> at worst 2-way conflicts.
        #pragma unroll
        for (int k = 0; k < 4; ++k)
            tile[(p0 + k) * NCH + c] = r[k];
    }
    __syncthreads();   // orders all waves' ds stores before the async copy-out

    // ---- Phase 2: pure linear async DMA, LDS -> global ---------------------
    // dst tile byte offset is a multiple of 16 (229840*4 and 8840*4 both are).
    float* dst = out + (size_t)slice * ((size_t)SPATIAL * NCH) + (size_t)s0 * NCH;
    const uint32_t lds_base = (uint32_t)(uintptr_t)tile;   // low 32 bits = LDS offset
    for (int i = tid; i < NVEC; i += 256) {                // 9 iterations
        async_store_b128(dst + i * 4, lds_base + (uint32_t)i * 16u);
    }

    // Drain this wave's async stores (S_ENDPGM also implies wait-idle).
#if __has_builtin(__builtin_amdgcn_s_wait_asynccnt)
    __builtin_amdgcn_s_wait_asynccnt(0);
#else
    asm volatile("s_wait_asynccnt 0" ::: "memory");
#endif
}

extern "C" void kernel_launch(void* const* d_in, const int* in_sizes, int n_in,
                              void* d_out, int out_size, void* d_ws, size_t ws_size,
                              hipStream_t stream) {
    (void)in_sizes; (void)n_in; (void)out_size; (void)d_ws; (void)ws_size;
    const float* in  = (const float*)d_in[0];
    float*       out = (float*)d_out;

    dim3 grid(NSLICES * TILES_PER_SLICE);   // 96 * 26 = 2496 workgroups
    dim3 block(256);                        // 8 wave32s per workgroup
    yolo_decode_kernel<<<grid, block, 0, stream>>>(in, out);
}